// GraphAttentionLayer_781684048710
// MI455X (gfx1250) — compile-verified
//
#include <hip/hip_runtime.h>
#include <hip/hip_bf16.h>

#define N_NODES 4096
#define E_EDGES 131072
#define IN_FEATS 256
#define OUT_FEATS 64
#define H_HEADS 8
#define WH_COLS (OUT_FEATS * H_HEADS)   // 512

typedef __attribute__((ext_vector_type(2))) float v2f;
typedef __attribute__((ext_vector_type(8))) float v8f;

// ---------------- workspace layout (float offsets) ----------------
#define WH_OFF     0                       // 2,097,152 floats (Wh: N x 512)
#define EDGEE_OFF  2097152                 // 1,048,576 floats (edge logits E x H)
#define AS_OFF     3145728                 // 32,768 (alpha_src N x H)
#define AD_OFF     3178496                 // 32,768 (alpha_dst N x H)
#define RMAX_OFF   3211264                 // 32,768 (row max, init 0)
#define RSUM_OFF   3244032                 // 32,768 (row sum exp)
#define INVZ_OFF   3276800                 // 32,768
#define BG_OFF     3309568                 // 32,768 (exp(-max)/Z)
#define S_OFF      3342336                 // 512 (per-column sums of Wh)
#define CNT_OFF    3342848                 // 4,096 ints (out-degree)
#define WS_FLOATS  3346944

// ---------------- Wh = x @ W via V_WMMA_F32_16X16X4_F32 ----------------
// One wave per 16x16 tile of the 4096x512 output; K loop over 256 in steps of 4.
__global__ __launch_bounds__(32) void gemm_wh_kernel(const float* __restrict__ x,
                                                     const float* __restrict__ W,
                                                     float* __restrict__ Wh) {
  const int lane = threadIdx.x;        // 0..31, wave32
  const int tn   = blockIdx.x;         // 0..31  (512/16)
  const int tm   = blockIdx.y;         // 0..255 (4096/16)
  const int lo   = lane & 15;
  const int hi   = lane >> 4;          // 0 or 1
  const int kb   = hi * 2;             // A/B K sub-block per half-wave
  const int m    = tm * 16 + lo;       // A row this lane feeds
  const int col  = tn * 16 + lo;       // B/D column this lane feeds

  v8f acc = {};
  const float* xrow = x + m * IN_FEATS;
  for (int k0 = 0; k0 < IN_FEATS; k0 += 4) {
    // A 16x4 f32: lanes 0-15 hold K=k0+0/1, lanes 16-31 hold K=k0+2/3
    const float2 av = *(const float2*)(xrow + k0 + kb);
    v2f a; a.x = av.x; a.y = av.y;
    // B 4x16 f32: vgpr v holds row (kb+v), N = lane%16
    v2f b;
    b.x = W[(k0 + kb + 0) * WH_COLS + col];
    b.y = W[(k0 + kb + 1) * WH_COLS + col];
    acc = __builtin_amdgcn_wmma_f32_16x16x4_f32(false, a, false, b,
                                                (short)0, acc, false, false);
  }
  // D 16x16 f32: vgpr v -> row v (lanes 0-15) / v+8 (lanes 16-31), col = lane%16
  const int row0 = tm * 16 + hi * 8;
#pragma unroll
  for (int v = 0; v < 8; ++v)
    Wh[(row0 + v) * WH_COLS + col] = acc[v];
}

// ---------------- alpha_src / alpha_dst : (N,H,64) . a1/a2 ----------------
__global__ void alpha_kernel(const float* __restrict__ Wh, const float* __restrict__ a,
                             float* __restrict__ as_, float* __restrict__ ad_) {
  int i = blockIdx.x * blockDim.x + threadIdx.x;    // n*H + h
  if (i >= N_NODES * H_HEADS) return;
  int n = i >> 3, h = i & 7;
  const float* wrow = Wh + n * WH_COLS + h * OUT_FEATS;
  float s1 = 0.f, s2 = 0.f;
#pragma unroll 8
  for (int f = 0; f < OUT_FEATS; ++f) {
    float w = wrow[f];
    s1 += w * a[f];
    s2 += w * a[OUT_FEATS + f];
  }
  as_[i] = s1; ad_[i] = s2;
}

// ---------------- S[c] = sum_n Wh[n][c] ----------------
__global__ void colsum_kernel(const float* __restrict__ Wh, float* __restrict__ S) {
  int r0 = blockIdx.x * 256;                        // 16 blocks x 256 rows
  for (int c = threadIdx.x; c < WH_COLS; c += blockDim.x) {
    float s = 0.f;
    for (int r = 0; r < 256; ++r) s += Wh[(r0 + r) * WH_COLS + c];
    atomicAdd(&S[c], s);
  }
}

// ---------------- pass 1: edge logits + per-(src,h) running max, degree ----------------
__global__ void edge_max_kernel(const int* __restrict__ ei,
                                const float* __restrict__ as_, const float* __restrict__ ad_,
                                float* __restrict__ edge_e, float* __restrict__ rowmax,
                                int* __restrict__ cnt) {
  int e = blockIdx.x * blockDim.x + threadIdx.x;
  if (e >= E_EDGES) return;
  int s = ei[e], d = ei[E_EDGES + e];
  atomicAdd(&cnt[s], 1);
#pragma unroll
  for (int h = 0; h < H_HEADS; ++h) {
    float z  = as_[s * H_HEADS + h] + ad_[d * H_HEADS + h];
    float ev = z > 0.f ? z : 0.2f * z;              // leaky_relu slope 0.2
    edge_e[e * H_HEADS + h] = ev;
    // rowmax init 0.0f; for candidates vs a nonnegative current value,
    // IEEE-float-as-signed-int ordering is exact -> int atomicMax is valid.
    atomicMax((int*)&rowmax[s * H_HEADS + h], __float_as_int(ev));
  }
}

// ---------------- pass 2: per-(src,h) sum of exp(e - max) ----------------
__global__ void edge_sum_kernel(const int* __restrict__ ei,
                                const float* __restrict__ edge_e,
                                const float* __restrict__ rowmax,
                                float* __restrict__ rowsum) {
  int e = blockIdx.x * blockDim.x + threadIdx.x;
  if (e >= E_EDGES) return;
  int s = ei[e];
#pragma unroll
  for (int h = 0; h < H_HEADS; ++h) {
    int i = s * H_HEADS + h;
    atomicAdd(&rowsum[i], expf(edge_e[e * H_HEADS + h] - rowmax[i]));
  }
}

// ---------------- finalize: Z = (N - deg)exp(-max) + rowsum ----------------
__global__ void finalize_kernel(const float* __restrict__ rowmax,
                                const float* __restrict__ rowsum,
                                const int* __restrict__ cnt,
                                float* __restrict__ invZ, float* __restrict__ bg) {
  int i = blockIdx.x * blockDim.x + threadIdx.x;
  if (i >= N_NODES * H_HEADS) return;
  int n = i >> 3;
  float enm = expf(-rowmax[i]);
  float Z   = (float)(N_NODES - cnt[n]) * enm + rowsum[i];
  float iz  = 1.0f / Z;
  invZ[i] = iz;
  bg[i]   = enm * iz;
}

// ---------------- out = bg * S (background softmax mass over all nodes) ----------------
__global__ void out_init_kernel(const float* __restrict__ S, const float* __restrict__ bg,
                                float* __restrict__ out) {
  int i = blockIdx.x * blockDim.x + threadIdx.x;
  if (i >= N_NODES * WH_COLS) return;
  int n = i / WH_COLS, c = i % WH_COLS, h = c >> 6;
  out[i] = bg[n * H_HEADS + h] * S[c];
}

// ---------------- per-edge correction scatter: out[s] += (softmax(e)-bg)*Wh[d] ----------------
__global__ void edge_scatter_kernel(const int* __restrict__ ei,
                                    const float* __restrict__ edge_e,
                                    const float* __restrict__ rowmax,
                                    const float* __restrict__ invZ,
                                    const float* __restrict__ bg,
                                    const float* __restrict__ Wh,
                                    float* __restrict__ out) {
  int t = blockIdx.x * blockDim.x + threadIdx.x;    // E * 64 threads
  if (t >= E_EDGES * OUT_FEATS) return;
  int e = t >> 6, f = t & 63;
  int s = ei[e], d = ei[E_EDGES + e];
  const float* wd = Wh + d * WH_COLS;
  float* os = out + s * WH_COLS;
#pragma unroll
  for (int h = 0; h < H_HEADS; ++h) {
    int i = s * H_HEADS + h;
    float coef = expf(edge_e[e * H_HEADS + h] - rowmax[i]) * invZ[i] - bg[i];
    atomicAdd(&os[h * OUT_FEATS + f], coef * wd[h * OUT_FEATS + f]);
  }
}

extern "C" void kernel_launch(void* const* d_in, const int* in_sizes, int n_in,
                              void* d_out, int out_size, void* d_ws, size_t ws_size,
                              hipStream_t stream) {
  (void)in_sizes; (void)n_in; (void)out_size;
  if (ws_size < (size_t)WS_FLOATS * sizeof(float)) return;

  const float* x  = (const float*)d_in[0];
  const float* W  = (const float*)d_in[1];
  const float* a  = (const float*)d_in[2];
  const int*   ei = (const int*)d_in[3];
  float* out = (float*)d_out;
  float* ws  = (float*)d_ws;

  float* Wh     = ws + WH_OFF;
  float* edge_e = ws + EDGEE_OFF;
  float* as_    = ws + AS_OFF;
  float* ad_    = ws + AD_OFF;
  float* rowmax = ws + RMAX_OFF;
  float* rowsum = ws + RSUM_OFF;
  float* invZ   = ws + INVZ_OFF;
  float* bg     = ws + BG_OFF;
  float* S      = ws + S_OFF;
  int*   cnt    = (int*)(ws + CNT_OFF);

  // zero accumulators (rowmax|rowsum contiguous; S|cnt contiguous)
  hipMemsetAsync(rowmax, 0, 2 * N_NODES * H_HEADS * sizeof(float), stream);
  hipMemsetAsync(S, 0, (WH_COLS + N_NODES) * sizeof(float), stream);

  gemm_wh_kernel<<<dim3(WH_COLS / 16, N_NODES / 16), 32, 0, stream>>>(x, W, Wh);
  alpha_kernel<<<(N_NODES * H_HEADS + 255) / 256, 256, 0, stream>>>(Wh, a, as_, ad_);
  colsum_kernel<<<N_NODES / 256, 256, 0, stream>>>(Wh, S);
  edge_max_kernel<<<E_EDGES / 256, 256, 0, stream>>>(ei, as_, ad_, edge_e, rowmax, cnt);
  edge_sum_kernel<<<E_EDGES / 256, 256, 0, stream>>>(ei, edge_e, rowmax, rowsum);
  finalize_kernel<<<(N_NODES * H_HEADS + 255) / 256, 256, 0, stream>>>(rowmax, rowsum, cnt, invZ, bg);
  out_init_kernel<<<(N_NODES * WH_COLS + 255) / 256, 256, 0, stream>>>(S, bg, out);
  edge_scatter_kernel<<<(E_EDGES * OUT_FEATS) / 256, 256, 0, stream>>>(
      ei, edge_e, rowmax, invZ, bg, Wh, out);
}